// TimeAwareMultiHeadAttention_4406636446375
// MI455X (gfx1250) — compile-verified
//
#include <hip/hip_runtime.h>

typedef float v2f __attribute__((ext_vector_type(2)));
typedef float v4f __attribute__((ext_vector_type(4)));
typedef float v8f __attribute__((ext_vector_type(8)));

#define B_  8
#define L_  200
#define H_  128
#define NH_ 4
#define HD_ 32
#define NEGF (-4294967295.0f)          // -(2^32)+1 from the reference
#define QSCALE 0.17677669529663687f    // 1/sqrt(32)

// ---------------------------------------------------------------------------
// Kernel A: fused projections via f32 WMMA.
//   which=0: Qp = (queries @ Wq.T + bq) * (1/sqrt(HD))
//   which=1: Kp = (keys    @ Wk.T + bk) + abs_pos_K
//   which=2: Vp = (keys    @ Wv.T + bv) + abs_pos_V
// One wave computes one 16x16 output tile using V_WMMA_F32_16X16X4_F32,
// looping K=128 in steps of 4 (32 WMMAs per tile).
// f32 A-frag layout (16x4): lane l holds A[l&15, 2*(l>>4)] , A[l&15, 2*(l>>4)+1]
// f32 B-frag layout (4x16): lane l holds B[2*(l>>4), l&15] , B[2*(l>>4)+1, l&15]
//   (B is K x N with B[k,n] = W[n,k], so both frags are contiguous float2 loads)
// C/D layout: reg r -> row r (lanes 0-15) / row r+8 (lanes 16-31), col = lane&15
// ---------------------------------------------------------------------------
__global__ __launch_bounds__(32) void taa_proj_wmma(
    const float* __restrict__ queries, const float* __restrict__ keys,
    const float* __restrict__ Wq, const float* __restrict__ bq,
    const float* __restrict__ Wk, const float* __restrict__ bk,
    const float* __restrict__ Wv, const float* __restrict__ bv,
    const float* __restrict__ pK, const float* __restrict__ pV,
    float* __restrict__ Qp, float* __restrict__ Kp, float* __restrict__ Vp)
{
    const int ROWS = B_ * L_;        // 1600
    const int MT   = ROWS / 16;      // 100
    const int NT   = H_ / 16;        // 8

    int bid   = blockIdx.x;
    int which = bid / (MT * NT);
    int rem   = bid % (MT * NT);
    int m0    = (rem / NT) * 16;
    int n0    = (rem % NT) * 16;

    const float* X;  const float* W;  const float* bias;
    if (which == 0)      { X = queries; W = Wq; bias = bq; }
    else if (which == 1) { X = keys;    W = Wk; bias = bk; }
    else                 { X = keys;    W = Wv; bias = bv; }

    const int lane  = threadIdx.x;         // 0..31
    const int mr    = lane & 15;
    const int khalf = (lane >> 4) * 2;     // 0 or 2

    const float* xrow = X + (size_t)(m0 + mr) * H_;   // A rows
    const float* wrow = W + (size_t)(n0 + mr) * H_;   // B cols (= W rows)

    v8f acc = {};
    #pragma unroll 4
    for (int k0 = 0; k0 < H_; k0 += 4) {
        v2f a = *(const v2f*)(xrow + k0 + khalf);
        v2f b = *(const v2f*)(wrow + k0 + khalf);
        acc = __builtin_amdgcn_wmma_f32_16x16x4_f32(
                  false, a, false, b, (short)0, acc, false, false);
    }

    const int n      = n0 + (lane & 15);
    const int rowadd = (lane >> 4) * 8;
    const float bn   = bias[n];

    #pragma unroll
    for (int r = 0; r < 8; ++r) {
        int m = m0 + r + rowadd;
        float v = acc[r] + bn;
        size_t idx = (size_t)m * H_ + n;
        if (which == 0)      Qp[idx] = v * QSCALE;
        else if (which == 1) Kp[idx] = v + pK[idx];
        else                 Vp[idx] = v + pV[idx];
    }
}

// ---------------------------------------------------------------------------
// Kernel B: fused score + softmax + weighted sum. One block per (b,q) row.
// 8 waves; lane l owns the float4 feature slice d = 4l..4l+3, head = l>>3.
// Phase 1: scores[h][k] = Q . (K' + tK + dK)  (masked -> NEG, loads skipped)
// Phase 2: per-head softmax (4 waves).
// Phase 3: out = sum_k p[h,k] * (V' + tV + dV)  (p==0 -> loads skipped)
// Big-tensor reads are non-temporal so K'/V' stay L2-resident.
// ---------------------------------------------------------------------------
__global__ __launch_bounds__(256) void taa_attn_fused(
    const float* __restrict__ Qp, const float* __restrict__ Kp,
    const float* __restrict__ Vp, const int* __restrict__ time_mask,
    const float* __restrict__ tK, const float* __restrict__ tV,
    const float* __restrict__ dK, const float* __restrict__ dV,
    float* __restrict__ out)
{
    __shared__ float s_scores[NH_][L_];
    __shared__ float s_part[8][H_];

    const int bq   = blockIdx.x;        // b*L + q
    const int b    = bq / L_;
    const int q    = bq - b * L_;
    const int tid  = threadIdx.x;
    const int lane = tid & 31;
    const int wave = tid >> 5;
    const int head = lane >> 3;

    const bool row_masked = (time_mask[bq] != 0);

    const v4f q4 = *(const v4f*)(Qp + (size_t)bq * H_ + 4 * lane);

    const size_t rowbase = (size_t)bq * L_ * H_;     // [b,q,:,:] slab
    const float* tKrow = tK + rowbase;
    const float* dKrow = dK + rowbase;

    // ---- Phase 1: scores ----
    for (int k = wave; k < L_; k += 8) {
        float score;
        if (row_masked || (k > q)) {
            score = NEGF;
        } else {
            size_t off = (size_t)k * H_ + 4 * lane;
            v4f kp = *(const v4f*)(Kp + (size_t)(b * L_ + k) * H_ + 4 * lane);
            v4f tk = __builtin_nontemporal_load((const v4f*)(tKrow + off));
            v4f dk = __builtin_nontemporal_load((const v4f*)(dKrow + off));
            v4f s  = (kp + tk + dk) * q4;
            float p = s.x + s.y + s.z + s.w;
            p += __shfl_xor(p, 1);
            p += __shfl_xor(p, 2);
            p += __shfl_xor(p, 4);
            score = p;
        }
        if ((lane & 7) == 0) s_scores[head][k] = score;
    }
    __syncthreads();

    // ---- Phase 2: softmax per head (waves 0..3) ----
    if (wave < NH_) {
        const int h = wave;
        float vals[7];
        float mx = -__builtin_inff();
        #pragma unroll
        for (int i = 0; i < 7; ++i) {
            int k = lane + 32 * i;
            float v = (k < L_) ? s_scores[h][k] : -__builtin_inff();
            vals[i] = v;
            mx = fmaxf(mx, v);
        }
        #pragma unroll
        for (int off = 1; off < 32; off <<= 1) mx = fmaxf(mx, __shfl_xor(mx, off));

        float sum = 0.0f;
        #pragma unroll
        for (int i = 0; i < 7; ++i) {
            int k = lane + 32 * i;
            if (k < L_) {
                float e = __expf(vals[i] - mx);   // fully-masked row -> exp(0)=1 -> uniform
                s_scores[h][k] = e;
                sum += e;
            }
        }
        #pragma unroll
        for (int off = 1; off < 32; off <<= 1) sum += __shfl_xor(sum, off);

        const float inv = 1.0f / sum;
        #pragma unroll
        for (int i = 0; i < 7; ++i) {
            int k = lane + 32 * i;
            if (k < L_) s_scores[h][k] *= inv;
        }
    }
    __syncthreads();

    // ---- Phase 3: weighted accumulation ----
    const float* tVrow = tV + rowbase;
    const float* dVrow = dV + rowbase;
    v4f acc = {};
    for (int k = wave; k < L_; k += 8) {
        float p = s_scores[head][k];
        if (p != 0.0f) {                 // causally-masked probs are exactly 0
            size_t off = (size_t)k * H_ + 4 * lane;
            v4f vp = *(const v4f*)(Vp + (size_t)(b * L_ + k) * H_ + 4 * lane);
            v4f tv = __builtin_nontemporal_load((const v4f*)(tVrow + off));
            v4f dv = __builtin_nontemporal_load((const v4f*)(dVrow + off));
            acc += p * (vp + tv + dv);
        }
    }
    *(v4f*)(&s_part[wave][4 * lane]) = acc;
    __syncthreads();

    if (tid < H_) {
        float s = 0.0f;
        #pragma unroll
        for (int w = 0; w < 8; ++w) s += s_part[w][tid];
        out[(size_t)bq * H_ + tid] = s;
    }
}

// ---------------------------------------------------------------------------
extern "C" void kernel_launch(void* const* d_in, const int* in_sizes, int n_in,
                              void* d_out, int out_size, void* d_ws, size_t ws_size,
                              hipStream_t stream) {
    const float* queries   = (const float*)d_in[0];
    const float* keys      = (const float*)d_in[1];
    const int*   time_mask = (const int*)  d_in[2];
    // d_in[3] attn_mask is analytic triu(k=1): computed as (k > q), never loaded
    const float* tK = (const float*)d_in[4];
    const float* tV = (const float*)d_in[5];
    const float* dK = (const float*)d_in[6];
    const float* dV = (const float*)d_in[7];
    const float* pK = (const float*)d_in[8];
    const float* pV = (const float*)d_in[9];
    const float* Wq = (const float*)d_in[10];
    const float* bq = (const float*)d_in[11];
    const float* Wk = (const float*)d_in[12];
    const float* bk = (const float*)d_in[13];
    const float* Wv = (const float*)d_in[14];
    const float* bv = (const float*)d_in[15];

    float* Qp = (float*)d_ws;                 // [B,L,H], pre-scaled by 1/sqrt(HD)
    float* Kp = Qp + (size_t)B_ * L_ * H_;    // [B,L,H], includes +abs_pos_K
    float* Vp = Kp + (size_t)B_ * L_ * H_;    // [B,L,H], includes +abs_pos_V

    const int ROWS = B_ * L_;                            // 1600
    const int tilesA = 3 * (ROWS / 16) * (H_ / 16);      // 2400 waves
    taa_proj_wmma<<<tilesA, 32, 0, stream>>>(
        queries, keys, Wq, bq, Wk, bk, Wv, bv, pK, pV, Qp, Kp, Vp);

    taa_attn_fused<<<B_ * L_, 256, 0, stream>>>(
        Qp, Kp, Vp, time_mask, tK, tV, dK, dV, (float*)d_out);
}